// GNN_graphpred_45011257262539
// MI455X (gfx1250) — compile-verified
//
#include <hip/hip_runtime.h>

typedef __attribute__((ext_vector_type(2))) float v2f;
typedef __attribute__((ext_vector_type(8))) float v8f;

#define EMB     128
#define NODE_F  32
#define EDGE_F  16
#define HID     256
#define NGRAPHS 64
#define NLAYER  3
#define BN_EPS  1e-5f

// ---------------------------------------------------------------------------
// h[N,128] = x[N,32] @ W0^T + b0         (f32 WMMA, 16x16 tiles, K steps of 4)
// Fragment layouts per CDNA5 ISA 7.12.2:
//   A 16x4 f32 : lane(mr=l&15) row m0+mr ; VGPR0 = K=(2*half), VGPR1 = K=(2*half+1)
//   B 4x16 f32 : lane col n0+mr          ; VGPR0 = K=(2*half), VGPR1 = K=(2*half+1)
//   C/D 16x16  : c[r] -> row m0 + r + 8*half, col n0+mr
// ---------------------------------------------------------------------------
__global__ __launch_bounds__(256) void k_node_embed(
    const float* __restrict__ x, const float* __restrict__ W0,
    const float* __restrict__ b0, float* __restrict__ h, int N)
{
    const int wave = threadIdx.x >> 5;        // 8 waves -> 8 column tiles of 16
    const int lane = threadIdx.x & 31;
    const int half = lane >> 4, mr = lane & 15;
    const int m0 = blockIdx.x * 16;
    const int n0 = wave * 16;

    v8f c = {};
    int row = m0 + mr; if (row >= N) row = N - 1;     // clamp; stores guarded
    #pragma unroll
    for (int k0 = 0; k0 < NODE_F; k0 += 4) {
        const int k = k0 + 2 * half;
        v2f a; a.x = x[(size_t)row * NODE_F + k];
               a.y = x[(size_t)row * NODE_F + k + 1];
        v2f b; b.x = W0[(size_t)(n0 + mr) * NODE_F + k];
               b.y = W0[(size_t)(n0 + mr) * NODE_F + k + 1];
        c = __builtin_amdgcn_wmma_f32_16x16x4_f32(false, a, false, b,
                                                  (short)0, c, false, false);
    }
    const float bias = b0[n0 + mr];
    if (m0 + 16 <= N) {                               // full tile: straight-line
        float* out = h + (size_t)(m0 + 8 * half) * EMB + n0 + mr;
        #pragma unroll
        for (int r = 0; r < 8; ++r) out[(size_t)r * EMB] = c[r] + bias;
    } else {
        #pragma unroll
        for (int r = 0; r < 8; ++r) {
            const int orow = m0 + r + 8 * half;
            if (orow < N) h[(size_t)orow * EMB + n0 + mr] = c[r] + bias;
        }
    }
}

// ---------------------------------------------------------------------------
// aggr[i] = h[i] + be (self-loop message), float4-vectorized.
// First 256 lanes also zero the BN stats accumulator.
// ---------------------------------------------------------------------------
__global__ void k_init_aggr(const float4* __restrict__ h4,
                            const float4* __restrict__ be4,
                            float4* __restrict__ aggr4,
                            float* __restrict__ stats, long total4)
{
    const long i = (long)blockIdx.x * blockDim.x + threadIdx.x;
    if (i < 2 * EMB) stats[i] = 0.0f;
    if (i >= total4) return;
    const int f4 = (int)(i & (EMB / 4 - 1));
    float4 hv = h4[i], bv = be4[f4];
    hv.x += bv.x; hv.y += bv.y; hv.z += bv.z; hv.w += bv.w;
    aggr4[i] = hv;
}

// ---------------------------------------------------------------------------
// One wave per edge: aggr[dst] += h[src] + We @ edge_attr[e] + be
// (edge-embedding GEMV fused into the scatter; 4 features per lane)
// ---------------------------------------------------------------------------
__global__ __launch_bounds__(256) void k_edge_msgs(
    const float* __restrict__ h, const float* __restrict__ ea,
    const float* __restrict__ We_l, const float* __restrict__ be_l,
    const int* __restrict__ src, const int* __restrict__ dst,
    float* __restrict__ aggr, int E)
{
    __shared__ float sWe[EMB * EDGE_F];
    __shared__ float sbe[EMB];
    for (int i = threadIdx.x; i < EMB * EDGE_F; i += 256) sWe[i] = We_l[i];
    if (threadIdx.x < EMB) sbe[threadIdx.x] = be_l[threadIdx.x];
    __syncthreads();

    const int wid  = (int)(((long)blockIdx.x * blockDim.x + threadIdx.x) >> 5);
    const int lane = threadIdx.x & 31;
    if (wid >= E) return;

    const int s = src[wid], d = dst[wid];
    const int f0 = lane * 4;
    const float4 hv = *(const float4*)(h + (size_t)s * EMB + f0);

    alignas(16) float eav[EDGE_F];
    const float4* ea4 = (const float4*)(ea + (size_t)wid * EDGE_F);
    #pragma unroll
    for (int q = 0; q < EDGE_F / 4; ++q) ((float4*)eav)[q] = ea4[q];

    float m[4];
    #pragma unroll
    for (int j = 0; j < 4; ++j) {
        const float* w = sWe + (size_t)(f0 + j) * EDGE_F;
        float acc = sbe[f0 + j];
        #pragma unroll
        for (int k = 0; k < EDGE_F; ++k) acc = fmaf(w[k], eav[k], acc);
        m[j] = acc;
    }
    float* out = aggr + (size_t)d * EMB + f0;
    atomicAdd(out + 0, hv.x + m[0]);
    atomicAdd(out + 1, hv.y + m[1]);
    atomicAdd(out + 2, hv.z + m[2]);
    atomicAdd(out + 3, hv.w + m[3]);
}

// ---------------------------------------------------------------------------
// Fused MLP: h2 = relu(aggr @ W1^T + b1) @ W2^T + b2 for 32 rows per block.
// 2x2 accumulator tiling: each A/B fragment load feeds 2 WMMAs, halving
// weight (L2) traffic vs 16-row blocks. Intermediate [32,256] in padded LDS.
// Block-reduced BN statistics -> one atomic per feature per block.
// h2 may alias aggr (all aggr reads complete before the post-barrier stores).
// ---------------------------------------------------------------------------
#define TPAD (HID + 4)
__global__ __launch_bounds__(256) void k_mlp(
    const float* __restrict__ aggr,
    const float* __restrict__ W1, const float* __restrict__ b1v,
    const float* __restrict__ W2, const float* __restrict__ b2v,
    float* __restrict__ h2, float* __restrict__ stats, int N)
{
    __shared__ float t[32][TPAD];       // relu intermediate (33.3 KB)
    __shared__ float redS[2][EMB];      // per-half partial sums
    __shared__ float redQ[2][EMB];      // per-half partial sum-of-squares

    const int wave = threadIdx.x >> 5;
    const int lane = threadIdx.x & 31;
    const int half = lane >> 4, mr = lane & 15;
    const int m0 = blockIdx.x * 32;
    int row0 = m0 + mr;      if (row0 >= N) row0 = N - 1;
    int row1 = m0 + 16 + mr; if (row1 >= N) row1 = N - 1;

    // ---- GEMM1: [32,128] x [128,256] -> relu -> LDS.  Wave owns cols
    //      [n0, n0+32); 2 row-tiles x 2 col-tiles share fragment loads.
    {
        const int n0 = wave * 32;
        v8f c00 = {}, c01 = {}, c10 = {}, c11 = {};
        #pragma unroll 4
        for (int k0 = 0; k0 < EMB; k0 += 4) {
            const int k = k0 + 2 * half;
            v2f a0; a0.x = aggr[(size_t)row0 * EMB + k];
                    a0.y = aggr[(size_t)row0 * EMB + k + 1];
            v2f a1; a1.x = aggr[(size_t)row1 * EMB + k];
                    a1.y = aggr[(size_t)row1 * EMB + k + 1];
            v2f bA; bA.x = W1[(size_t)(n0 + mr) * EMB + k];
                    bA.y = W1[(size_t)(n0 + mr) * EMB + k + 1];
            v2f bB; bB.x = W1[(size_t)(n0 + 16 + mr) * EMB + k];
                    bB.y = W1[(size_t)(n0 + 16 + mr) * EMB + k + 1];
            c00 = __builtin_amdgcn_wmma_f32_16x16x4_f32(false, a0, false, bA, (short)0, c00, false, false);
            c10 = __builtin_amdgcn_wmma_f32_16x16x4_f32(false, a1, false, bA, (short)0, c10, false, false);
            c01 = __builtin_amdgcn_wmma_f32_16x16x4_f32(false, a0, false, bB, (short)0, c01, false, false);
            c11 = __builtin_amdgcn_wmma_f32_16x16x4_f32(false, a1, false, bB, (short)0, c11, false, false);
        }
        const float biasA = b1v[n0 + mr];
        const float biasB = b1v[n0 + 16 + mr];
        #pragma unroll
        for (int r = 0; r < 8; ++r) {
            const int lr = r + 8 * half;
            float v;
            v = c00[r] + biasA; t[lr     ][n0 + mr     ] = v > 0.0f ? v : 0.0f;
            v = c01[r] + biasB; t[lr     ][n0 + 16 + mr] = v > 0.0f ? v : 0.0f;
            v = c10[r] + biasA; t[lr + 16][n0 + mr     ] = v > 0.0f ? v : 0.0f;
            v = c11[r] + biasB; t[lr + 16][n0 + 16 + mr] = v > 0.0f ? v : 0.0f;
        }
    }
    __syncthreads();

    // ---- GEMM2: [32,256](LDS) x [256,128] -> h2.  Wave owns cols [p0,p0+16);
    //      B fragment shared across both row-tiles.
    {
        const int p0 = wave * 16;
        v8f d0 = {}, d1 = {};
        #pragma unroll 4
        for (int k0 = 0; k0 < HID; k0 += 4) {
            const int k = k0 + 2 * half;
            v2f b;  b.x  = W2[(size_t)(p0 + mr) * HID + k];
                    b.y  = W2[(size_t)(p0 + mr) * HID + k + 1];
            v2f a0; a0.x = t[mr][k];      a0.y = t[mr][k + 1];
            v2f a1; a1.x = t[16 + mr][k]; a1.y = t[16 + mr][k + 1];
            d0 = __builtin_amdgcn_wmma_f32_16x16x4_f32(false, a0, false, b, (short)0, d0, false, false);
            d1 = __builtin_amdgcn_wmma_f32_16x16x4_f32(false, a1, false, b, (short)0, d1, false, false);
        }
        const float bias = b2v[p0 + mr];
        float psum = 0.0f, psq = 0.0f;
        if (m0 + 32 <= N) {                           // full tile: straight-line
            float* out0 = h2 + (size_t)(m0 + 8 * half) * EMB + p0 + mr;
            #pragma unroll
            for (int r = 0; r < 8; ++r) {
                float v0 = d0[r] + bias;
                float v1 = d1[r] + bias;
                out0[(size_t)r * EMB]        = v0;
                out0[(size_t)(r + 16) * EMB] = v1;
                psum += v0 + v1;
                psq  += v0 * v0 + v1 * v1;
            }
        } else {
            #pragma unroll
            for (int r = 0; r < 8; ++r) {
                const int ra = m0 + r + 8 * half;
                const int rb = ra + 16;
                float v0 = d0[r] + bias;
                float v1 = d1[r] + bias;
                if (ra < N) { h2[(size_t)ra * EMB + p0 + mr] = v0; psum += v0; psq += v0 * v0; }
                if (rb < N) { h2[(size_t)rb * EMB + p0 + mr] = v1; psum += v1; psq += v1 * v1; }
            }
        }
        redS[half][p0 + mr] = psum;
        redQ[half][p0 + mr] = psq;
    }
    __syncthreads();

    if (threadIdx.x < EMB) {
        const int f = threadIdx.x;
        atomicAdd(&stats[f],       redS[0][f] + redS[1][f]);
        atomicAdd(&stats[EMB + f], redQ[0][f] + redQ[1][f]);
    }
}

// ---------------------------------------------------------------------------
// BatchNorm apply (+optional relu), float4-vectorized:
//   h = gamma*(h2-mu)*rsqrt(var+eps)+beta
// ---------------------------------------------------------------------------
__global__ void k_bn(const float4* __restrict__ h24, const float* __restrict__ stats,
                     const float4* __restrict__ gamma4, const float4* __restrict__ beta4,
                     float4* __restrict__ h4, int N, long total4, int do_relu)
{
    const long i = (long)blockIdx.x * blockDim.x + threadIdx.x;
    if (i >= total4) return;
    const int f4 = (int)(i & (EMB / 4 - 1));
    const int f  = f4 * 4;
    const float invN = 1.0f / (float)N;
    const float4 g = gamma4[f4], bt = beta4[f4], v = h24[i];
    float4 o;
    {
        float mu = stats[f + 0] * invN, var = stats[EMB + f + 0] * invN - mu * mu;
        o.x = g.x * (v.x - mu) * rsqrtf(var + BN_EPS) + bt.x;
        mu = stats[f + 1] * invN; var = stats[EMB + f + 1] * invN - mu * mu;
        o.y = g.y * (v.y - mu) * rsqrtf(var + BN_EPS) + bt.y;
        mu = stats[f + 2] * invN; var = stats[EMB + f + 2] * invN - mu * mu;
        o.z = g.z * (v.z - mu) * rsqrtf(var + BN_EPS) + bt.z;
        mu = stats[f + 3] * invN; var = stats[EMB + f + 3] * invN - mu * mu;
        o.w = g.w * (v.w - mu) * rsqrtf(var + BN_EPS) + bt.w;
    }
    if (do_relu) {
        o.x = o.x > 0.0f ? o.x : 0.0f;
        o.y = o.y > 0.0f ? o.y : 0.0f;
        o.z = o.z > 0.0f ? o.z : 0.0f;
        o.w = o.w > 0.0f ? o.w : 0.0f;
    }
    h4[i] = o;
}

// ---------------------------------------------------------------------------
// Pooling + head
// ---------------------------------------------------------------------------
__global__ void k_zero_pool(float* __restrict__ pooled, float* __restrict__ cnt)
{
    const int i = blockIdx.x * blockDim.x + threadIdx.x;
    if (i < NGRAPHS * EMB) pooled[i] = 0.0f;
    if (i < NGRAPHS) cnt[i] = 0.0f;
}

__global__ void k_pool(const float4* __restrict__ h4, const int* __restrict__ batch,
                       float* __restrict__ pooled, float* __restrict__ cnt, long total4)
{
    const long i = (long)blockIdx.x * blockDim.x + threadIdx.x;
    if (i >= total4) return;
    const int node = (int)(i >> 5);               // 32 float4 per node
    const int f    = (int)(i & 31) * 4;
    const int g    = batch[node];
    const float4 v = h4[i];
    float* p = pooled + (size_t)g * EMB + f;
    atomicAdd(p + 0, v.x);
    atomicAdd(p + 1, v.y);
    atomicAdd(p + 2, v.z);
    atomicAdd(p + 3, v.w);
    if ((i & 31) == 0) atomicAdd(&cnt[g], 1.0f);
}

__global__ void k_head(const float* __restrict__ pooled, const float* __restrict__ cnt,
                       const float* __restrict__ Wp, const float* __restrict__ bp,
                       float* __restrict__ out)
{
    const int g = threadIdx.x;
    if (g >= NGRAPHS) return;
    float c = cnt[g]; c = c > 1.0f ? c : 1.0f;
    float acc = 0.0f;
    #pragma unroll 8
    for (int f = 0; f < EMB; ++f) acc = fmaf(pooled[(size_t)g * EMB + f], Wp[f], acc);
    out[g] = acc / c + bp[0];
}

// ---------------------------------------------------------------------------
extern "C" void kernel_launch(void* const* d_in, const int* in_sizes, int n_in,
                              void* d_out, int out_size, void* d_ws, size_t ws_size,
                              hipStream_t stream)
{
    const float* x         = (const float*)d_in[0];
    const float* edge_attr = (const float*)d_in[1];
    const float* W0        = (const float*)d_in[2];
    const float* b0        = (const float*)d_in[3];
    const float* We        = (const float*)d_in[4];
    const float* be        = (const float*)d_in[5];
    const float* W1        = (const float*)d_in[6];
    const float* b1        = (const float*)d_in[7];
    const float* W2        = (const float*)d_in[8];
    const float* b2        = (const float*)d_in[9];
    const float* gamma     = (const float*)d_in[10];
    const float* beta      = (const float*)d_in[11];
    const float* Wp        = (const float*)d_in[12];
    const float* bp        = (const float*)d_in[13];
    const int*   edge_index= (const int*)d_in[14];
    const int*   batch     = (const int*)d_in[15];

    const int N = in_sizes[0] / NODE_F;
    const int E = in_sizes[14] / 2;
    const int* src = edge_index;
    const int* dst = edge_index + E;

    float* ws = (float*)d_ws;
    size_t off = 0;
    float* h      = ws + off; off += (size_t)N * EMB;
    float* aggr   = ws + off; off += (size_t)N * EMB;
    float* h2     = aggr;                       // safe alias (see k_mlp note)
    float* stats  = ws + off; off += 2 * EMB;
    float* pooled = ws + off; off += NGRAPHS * EMB;
    float* cnt    = ws + off; off += NGRAPHS;

    const long total4  = (long)N * (EMB / 4);   // float4 element count
    const int  mtiles  = (N + 15) / 16;
    const int  mtiles2 = (N + 31) / 32;
    const int  vblocks = (int)((total4 + 255) / 256);
    const int  edgeblk = (E + 7) / 8;           // 1 edge per wave, 8 waves/block

    k_node_embed<<<mtiles, 256, 0, stream>>>(x, W0, b0, h, N);

    for (int l = 0; l < NLAYER; ++l) {
        k_init_aggr<<<vblocks, 256, 0, stream>>>((const float4*)h,
                                                 (const float4*)(be + (size_t)l * EMB),
                                                 (float4*)aggr, stats, total4);
        k_edge_msgs<<<edgeblk, 256, 0, stream>>>(h, edge_attr,
                                                 We + (size_t)l * EMB * EDGE_F,
                                                 be + (size_t)l * EMB,
                                                 src, dst, aggr, E);
        k_mlp<<<mtiles2, 256, 0, stream>>>(aggr,
                                           W1 + (size_t)l * HID * EMB, b1 + (size_t)l * HID,
                                           W2 + (size_t)l * EMB * HID, b2 + (size_t)l * EMB,
                                           h2, stats, N);
        k_bn<<<vblocks, 256, 0, stream>>>((const float4*)h2, stats,
                                          (const float4*)(gamma + (size_t)l * EMB),
                                          (const float4*)(beta + (size_t)l * EMB),
                                          (float4*)h, N, total4,
                                          (l < NLAYER - 1) ? 1 : 0);
    }

    k_zero_pool<<<(NGRAPHS * EMB + 255) / 256, 256, 0, stream>>>(pooled, cnt);
    k_pool<<<vblocks, 256, 0, stream>>>((const float4*)h, batch, pooled, cnt, total4);
    k_head<<<1, 64, 0, stream>>>(pooled, cnt, Wp, bp, (float*)d_out);
}